// OptimLayer_58042188038285
// MI455X (gfx1250) — compile-verified
//
#include <hip/hip_runtime.h>
#include <math.h>

// Problem constants (match reference)
#define BATCH     256
#define NSIZE     32
#define HID       128
#define MAX_ITERS 10
#define DAMPING   0.1f

typedef float v2f __attribute__((ext_vector_type(2)));
typedef float v8f __attribute__((ext_vector_type(8)));

// In-place 9x9 Cholesky (lower triangle), serial (run by one thread).
__device__ __forceinline__ void chol9(float* A) {
  for (int k = 0; k < 9; ++k) {
    float piv = sqrtf(A[k * 9 + k]);
    A[k * 9 + k] = piv;
    float inv = 1.0f / piv;
    for (int r = k + 1; r < 9; ++r) A[r * 9 + k] *= inv;
    for (int c = k + 1; c < 9; ++c)
      for (int r = c; r < 9; ++r)
        A[r * 9 + c] -= A[r * 9 + k] * A[c * 9 + k];
  }
}

// Map state component c (0..8) to MLP input column j (0..6) or -1.
// inp = [p_z, v0,v1,v2, w0,w1,w2, dt] ; state = [p(3), v(3), w(3)]
__device__ __forceinline__ int jcol(int c) {
  return (c == 2) ? 0 : ((c >= 3) ? (c - 2) : -1);
}

__global__ __launch_bounds__(256) void lm_block_tridiag_kernel(
    const float* __restrict__ x,    // (B, 32, 4)
    const float* __restrict__ w0,   // (B, 1, 3)
    const float* __restrict__ pwp,  // (3,)
    const float* __restrict__ wvwp, // (6,)
    const float* __restrict__ w0wp, // (3,)
    const float* __restrict__ W1,   // (8, 128)
    const float* __restrict__ b1,   // (128,)
    const float* __restrict__ W2,   // (128, 6)
    const float* __restrict__ b2,   // (6,)
    float* __restrict__ out)        // p0 (B*3) | v0 (B*3) | w0f (B*3)
{
  __shared__ float s[NSIZE * 9];        // state
  __shared__ float pprior[NSIZE * 3];
  __shared__ float dtv[NSIZE];
  __shared__ float wprior[3];
  __shared__ float pw_s[3], wvw_s[6], w0w_s[3];
  __shared__ float W1s[8 * HID];        // [j][k] : input j, hidden k
  __shared__ float b1s[HID];
  __shared__ float W2s[HID * 6];        // [k][o] : hidden k, output o
  __shared__ float b2s[6];
  __shared__ float inpL[NSIZE * 8];     // padded 32x8 (row 31 = 0)
  __shared__ float Hs[NSIZE * HID];     // tanh activations
  __shared__ float outs[NSIZE * 6];     // MLP outputs
  __shared__ float G[31 * 48];          // G[i][o][j]: d out / d inp (6x8 per step)
  __shared__ float Ablk[NSIZE * 81];    // diagonal 9x9 blocks of J^T J + damping
  __shared__ float Ublk[31 * 81];       // super-diagonal 9x9 blocks (then M_i)
  __shared__ float gv[NSIZE * 9];       // J^T r (then h_i)
  __shared__ float xv[NSIZE * 9];       // solution dx
  __shared__ float Ucopy[81];           // original U_i during elimination

  const int tid  = threadIdx.x;
  const int lane = tid & 31;
  const int wave = tid >> 5;
  const int blk  = blockIdx.x;

  const float* xb = x + blk * NSIZE * 4;

  // Per-lane WMMA fragment geometry (f32 16x16x4, wave32):
  //  A (16x4): lane L holds row M=L&15; lanes<16 hold K={k,k+1}, lanes>=16 K={k+2,k+3}
  //  B (4x16): lane L holds col N=L&15; same K split
  //  C/D (16x16): vgpr r holds row r + (L<16?0:8), col L&15
  const int koff  = (lane < 16) ? 0 : 2;
  const int cn    = lane & 15;
  const int rbase = (lane < 16) ? 0 : 8;

  // ---- Load inputs into LDS ----
  for (int i = tid; i < NSIZE * 3; i += 256) {
    int n = i / 3, a = i % 3;
    pprior[i] = xb[n * 4 + 1 + a];
  }
  for (int i = tid; i < NSIZE; i += 256)
    dtv[i] = (i < NSIZE - 1) ? (xb[(i + 1) * 4] - xb[i * 4]) : 0.0f;
  if (tid < 3) {
    wprior[tid] = w0[blk * 3 + tid];
    pw_s[tid]   = pwp[tid];
    w0w_s[tid]  = w0wp[tid];
  }
  if (tid < 6) { wvw_s[tid] = wvwp[tid]; b2s[tid] = b2[tid]; }
  for (int i = tid; i < 8 * HID; i += 256) W1s[i] = W1[i];
  for (int i = tid; i < HID; i += 256)     b1s[i] = b1[i];
  for (int i = tid; i < HID * 6; i += 256) W2s[i] = W2[i];
  // init state: p = pprior, v = 0, w = w0f
  for (int i = tid; i < NSIZE * 9; i += 256) {
    int n = i / 9, c = i % 9;
    float v;
    if (c < 3)      v = xb[n * 4 + 1 + c];
    else if (c < 6) v = 0.0f;
    else            v = w0[blk * 3 + (c - 6)];
    s[i] = v;
  }
  __syncthreads();

  for (int it = 0; it < MAX_ITERS; ++it) {
    // ---- Build MLP input (32x8 padded) ----
    for (int i = tid; i < NSIZE * 8; i += 256) {
      int n = i / 8, j = i % 8;
      float v = 0.0f;
      if (n < NSIZE - 1) v = (j < 7) ? s[n * 9 + 2 + j] : dtv[n];
      inpL[i] = v;
    }
    __syncthreads();

    // ---- Forward 1: H = tanh(inp @ W1 + b1) via WMMA f32 16x16x4 ----
    {
      #pragma unroll
      for (int rep = 0; rep < 2; ++rep) {
        int t  = wave + rep * 8;        // 16 tiles over 8 waves
        int mT = t & 1, nT = t >> 1;
        int m  = mT * 16 + cn;
        int n  = nT * 16 + cn;
        v8f acc = {0.f, 0.f, 0.f, 0.f, 0.f, 0.f, 0.f, 0.f};
        #pragma unroll
        for (int kk = 0; kk < 8; kk += 4) {
          v2f a, b;
          a.x = inpL[m * 8 + kk + koff];
          a.y = inpL[m * 8 + kk + koff + 1];
          b.x = W1s[(kk + koff) * HID + n];
          b.y = W1s[(kk + koff + 1) * HID + n];
          acc = __builtin_amdgcn_wmma_f32_16x16x4_f32(
              false, a, false, b, (short)0, acc, false, false);
        }
        #pragma unroll
        for (int r = 0; r < 8; ++r) {
          int row = mT * 16 + r + rbase;
          Hs[row * HID + n] = tanhf(acc[r] + b1s[n]);
        }
      }
    }
    __syncthreads();

    // ---- Forward 2 (waves 0..1): out = H @ W2 + b2 (N padded 6->16) ----
    if (wave < 2) {
      const int mT = wave;
      const int m  = mT * 16 + cn;
      v8f acc = {0.f, 0.f, 0.f, 0.f, 0.f, 0.f, 0.f, 0.f};
      for (int kk = 0; kk < HID; kk += 4) {
        v2f a, b;
        a.x = Hs[m * HID + kk + koff];
        a.y = Hs[m * HID + kk + koff + 1];
        b.x = (cn < 6) ? W2s[(kk + koff) * 6 + cn] : 0.0f;
        b.y = (cn < 6) ? W2s[(kk + koff + 1) * 6 + cn] : 0.0f;
        acc = __builtin_amdgcn_wmma_f32_16x16x4_f32(
            false, a, false, b, (short)0, acc, false, false);
      }
      if (cn < 6) {
        #pragma unroll
        for (int r = 0; r < 8; ++r) {
          int row = mT * 16 + r + rbase;
          outs[row * 6 + cn] = acc[r] + b2s[cn];
        }
      }
    }

    // ---- MLP Jacobians via WMMA (same barrier region; independent of outs):
    //      G_i = A @ B_i ; A[o][k] = W2[k][o] (6x128, pad M=16)
    //                      B_i[k][j] = (1-h_ik^2) * W1[j][k] (128x8, pad N=16)
    //      31 steps spread over 8 waves, one 16x16 tile each, 32 k-steps.
    #pragma unroll
    for (int rep = 0; rep < 4; ++rep) {
      int i = wave + rep * 8;           // wave-uniform
      if (i < NSIZE - 1) {
        const float* hrow = &Hs[i * HID];
        v8f acc = {0.f, 0.f, 0.f, 0.f, 0.f, 0.f, 0.f, 0.f};
        for (int kk = 0; kk < HID; kk += 4) {
          int k0 = kk + koff;
          v2f a, b;
          a.x = (cn < 6) ? W2s[k0 * 6 + cn] : 0.0f;
          a.y = (cn < 6) ? W2s[(k0 + 1) * 6 + cn] : 0.0f;
          float h0 = hrow[k0], h1 = hrow[k0 + 1];
          float d0 = 1.0f - h0 * h0, d1 = 1.0f - h1 * h1;
          b.x = (cn < 8) ? d0 * W1s[cn * HID + k0] : 0.0f;
          b.y = (cn < 8) ? d1 * W1s[cn * HID + k0 + 1] : 0.0f;
          acc = __builtin_amdgcn_wmma_f32_16x16x4_f32(
              false, a, false, b, (short)0, acc, false, false);
        }
        if (cn < 8) {
          #pragma unroll
          for (int r = 0; r < 8; ++r) {
            int o = r + rbase;
            if (o < 6) G[i * 48 + o * 8 + cn] = acc[r];
          }
        }
      }
    }
    __syncthreads();

    // ---- Assemble block-tridiagonal A = J^T J + damping*I (analytic) ----
    for (int idx = tid; idx < NSIZE * 81; idx += 256) {
      int i = idx / 81, e = idx % 81, c = e / 9, d = e % 9;
      float dt = dtv[i];
      float a = (c == d) ? DAMPING : 0.0f;
      if (c == d && c < 3) a += pw_s[c] * pw_s[c];                         // r_prior
      if (i < NSIZE - 1 && c < 6 && d < 6 && (c % 3) == (d % 3)) {          // r_pos_i
        float p2 = pw_s[c % 3] * pw_s[c % 3];
        a += p2 * ((c < 3) ? 1.0f : dt) * ((d < 3) ? 1.0f : dt);
      }
      if (i > 0 && c == d && c < 3) a += pw_s[c] * pw_s[c];                 // r_pos_{i-1}
      if (i > 0 && c == d && c >= 3) { float w = wvw_s[c - 3]; a += w * w; }// r_vw_{i-1}
      if (i == 0 && c == d && c >= 6) { float w = w0w_s[c - 6]; a += w * w; }// r_w0
      if (i < NSIZE - 1) {                                                  // r_vw_i: B^T B
        int jc = jcol(c), jd = jcol(d);
        if (jc >= 0 && jd >= 0) {
          const float* Gi = &G[i * 48];
          float acc = 0.0f;
          #pragma unroll
          for (int o = 0; o < 6; ++o) {
            float w = wvw_s[o];
            acc += w * w * Gi[o * 8 + jc] * Gi[o * 8 + jd];
          }
          a += acc;
        }
      }
      Ablk[idx] = a;
    }
    // super-diagonal blocks U_i (couples s_i rows to s_{i+1} cols)
    for (int idx = tid; idx < 31 * 81; idx += 256) {
      int i = idx / 81, e = idx % 81, c = e / 9, d = e % 9;
      float dt = dtv[i];
      float a = 0.0f;
      if (d < 3 && c < 6 && (c % 3) == d) {                                 // E^T F
        float p2 = pw_s[d] * pw_s[d];
        a -= p2 * ((c < 3) ? 1.0f : dt);
      }
      if (d >= 3) {                                                         // B^T C
        int jc = jcol(c);
        if (jc >= 0) { int o = d - 3; float w = wvw_s[o]; a -= w * w * G[i * 48 + o * 8 + jc]; }
      }
      Ublk[idx] = a;
    }
    // rhs g = J^T r
    for (int idx = tid; idx < NSIZE * 9; idx += 256) {
      int i = idx / 9, c = idx % 9;
      float a = 0.0f;
      if (c < 3) {
        float rp = pw_s[c] * (s[i * 9 + c] - pprior[i * 3 + c]);
        a += pw_s[c] * rp;
      }
      if (i < NSIZE - 1) {
        float dt = dtv[i];
        if (c < 6) {
          int ax = c % 3;
          float rpos = pw_s[ax] * (s[(i + 1) * 9 + ax] - s[i * 9 + ax] - s[i * 9 + 3 + ax] * dt);
          a -= pw_s[ax] * rpos * ((c < 3) ? 1.0f : dt);
        }
        int jc = jcol(c);
        if (jc >= 0) {
          float acc = 0.0f;
          #pragma unroll
          for (int o = 0; o < 6; ++o) {
            float rv = wvw_s[o] * (outs[i * 6 + o] - s[(i + 1) * 9 + 3 + o]);
            acc += wvw_s[o] * G[i * 48 + o * 8 + jc] * rv;
          }
          a += acc;
        }
      }
      if (i > 0) {
        float dtm = dtv[i - 1];
        if (c < 3) {
          float rpos = pw_s[c] * (s[i * 9 + c] - s[(i - 1) * 9 + c] - s[(i - 1) * 9 + 3 + c] * dtm);
          a += pw_s[c] * rpos;
        }
        if (c >= 3) {
          int o = c - 3;
          float rv = wvw_s[o] * (outs[(i - 1) * 6 + o] - s[i * 9 + 3 + o]);
          a -= wvw_s[o] * rv;
        }
      }
      if (i == 0 && c >= 6) {
        float rw = w0w_s[c - 6] * (s[c] - wprior[c - 6]);
        a += w0w_s[c - 6] * rw;
      }
      gv[idx] = a;
    }
    __syncthreads();

    // ---- Block-tridiagonal forward elimination ----
    for (int i = 0; i < NSIZE; ++i) {
      if (tid == 0) chol9(&Ablk[i * 81]);
      else if (i < NSIZE - 1 && tid >= 1 && tid < 82)
        Ucopy[tid - 1] = Ublk[i * 81 + tid - 1];
      __syncthreads();
      // Solve D_i Z = [U_i | g_i]  (column-parallel triangular solves)
      if ((tid < 9 && i < NSIZE - 1) || tid == 9) {
        float col[9];
        const float* L = &Ablk[i * 81];
        if (tid < 9) for (int r = 0; r < 9; ++r) col[r] = Ublk[i * 81 + r * 9 + tid];
        else         for (int r = 0; r < 9; ++r) col[r] = gv[i * 9 + r];
        for (int r = 0; r < 9; ++r) {               // L y = col
          float v = col[r];
          for (int k = 0; k < r; ++k) v -= L[r * 9 + k] * col[k];
          col[r] = v / L[r * 9 + r];
        }
        for (int r = 8; r >= 0; --r) {              // L^T z = y
          float v = col[r];
          for (int k = r + 1; k < 9; ++k) v -= L[k * 9 + r] * col[k];
          col[r] = v / L[r * 9 + r];
        }
        if (tid < 9) for (int r = 0; r < 9; ++r) Ublk[i * 81 + r * 9 + tid] = col[r]; // M_i
        else         for (int r = 0; r < 9; ++r) gv[i * 9 + r] = col[r];              // h_i
      }
      __syncthreads();
      // Schur update: D_{i+1} -= U_i^T M_i ; g_{i+1} -= U_i^T h_i
      if (i < NSIZE - 1) {
        if (tid < 81) {
          int r = tid / 9, c = tid % 9;
          float acc = 0.0f;
          #pragma unroll
          for (int k = 0; k < 9; ++k) acc += Ucopy[k * 9 + r] * Ublk[i * 81 + k * 9 + c];
          Ablk[(i + 1) * 81 + r * 9 + c] -= acc;
        } else if (tid < 90) {
          int c = tid - 81;
          float acc = 0.0f;
          #pragma unroll
          for (int k = 0; k < 9; ++k) acc += Ucopy[k * 9 + c] * gv[i * 9 + k];
          gv[(i + 1) * 9 + c] -= acc;
        }
      }
      __syncthreads();
    }

    // ---- Back substitution: x_i = h_i - M_i x_{i+1} ----
    if (tid < 9) xv[(NSIZE - 1) * 9 + tid] = gv[(NSIZE - 1) * 9 + tid];
    __syncthreads();
    for (int i = NSIZE - 2; i >= 0; --i) {
      if (tid < 9) {
        float acc = gv[i * 9 + tid];
        #pragma unroll
        for (int k = 0; k < 9; ++k) acc -= Ublk[i * 81 + tid * 9 + k] * xv[(i + 1) * 9 + k];
        xv[i * 9 + tid] = acc;
      }
      __syncthreads();
    }

    // ---- LM update: s <- s - dx ----
    for (int idx = tid; idx < NSIZE * 9; idx += 256) s[idx] -= xv[idx];
    __syncthreads();
  }

  // ---- Outputs: p0 | v0 | w0f ----
  if (tid < 3) {
    out[blk * 3 + tid]                 = s[tid];      // p0
    out[BATCH * 3 + blk * 3 + tid]     = s[3 + tid];  // v0
    out[2 * BATCH * 3 + blk * 3 + tid] = wprior[tid]; // w0f (pass-through)
  }
}

extern "C" void kernel_launch(void* const* d_in, const int* in_sizes, int n_in,
                              void* d_out, int out_size, void* d_ws, size_t ws_size,
                              hipStream_t stream) {
  (void)in_sizes; (void)n_in; (void)out_size; (void)d_ws; (void)ws_size;
  lm_block_tridiag_kernel<<<BATCH, 256, 0, stream>>>(
      (const float*)d_in[0], (const float*)d_in[1], (const float*)d_in[2],
      (const float*)d_in[3], (const float*)d_in[4], (const float*)d_in[5],
      (const float*)d_in[6], (const float*)d_in[7], (const float*)d_in[8],
      (float*)d_out);
}